// LSTM_584115552367
// MI455X (gfx1250) — compile-verified
//
#include <hip/hip_runtime.h>

// CDNA5 (gfx1250) wave32 LSTM with V_WMMA_F32_16X16X4_F32.
// One wave owns a 16-row batch tile. Weight B-fragments are hoisted into
// registers (loop-invariant); biases are folded into the K=17 padding row of
// the weights with a constant 1.0 in the K=17 slot of every A operand, so
// every WMMA chain starts from the inline-0 C operand.

typedef __attribute__((ext_vector_type(2))) float v2f;
typedef __attribute__((ext_vector_type(8))) float v8f;

#define B_TOT 65536
#define T_TOT 50
#define D_DIM 17
#define WPB   4            // waves per block
#define TPB   (WPB * 32)

__device__ __forceinline__ float sigm(float x) {
  return 1.0f / (1.0f + __expf(-x));
}
__device__ __forceinline__ float tanh_fast(float x) {
  return 2.0f / (1.0f + __expf(-2.0f * x)) - 1.0f;
}

__global__ __launch_bounds__(TPB) void lstm_wmma_kernel(
    const float* __restrict__ x,      // [B, T, D]
    const float* __restrict__ W_ih,   // [68, 17]
    const float* __restrict__ W_hh,   // [68, 17]
    const float* __restrict__ b_ih,   // [68]
    const float* __restrict__ b_hh,   // [68]
    const float* __restrict__ W_lin,  // [17, 17]
    const float* __restrict__ b_lin,  // [17]
    const float* __restrict__ mask_ih,
    const float* __restrict__ mask_hh,
    const int* __restrict__ slb_p,
    float* __restrict__ out)          // [B, T-SLB, 17]
{
  // Weights staged once as W[col][k] (masked), padded to k=20.
  // Row k=17 of sWih carries (b_ih+b_hh); row k=17 of sWlin carries b_lin.
  __shared__ __align__(16) float sWih[80][20];
  __shared__ __align__(16) float sWhh[80][20];
  __shared__ __align__(16) float sWlin[32][20];
  // Per-wave scratch. sGates stored [col][row]: each lane's 8 C values
  // (one column, rows m0..m0+7) go out as two ds_store_b128.
  __shared__ __align__(16) float sGates[WPB][80][20];
  __shared__ __align__(16) float sH[WPB][16][20];    // [row][k]; k17=1, k18/19=0
  __shared__ __align__(16) float sC[WPB][17][16];    // [d][row]
  __shared__ __align__(16) float sOut[WPB][16][20];  // [row][k]; k17=1, k18/19=0

  const int tid  = threadIdx.x;
  const int w    = tid >> 5;
  const int lane = tid & 31;
  const int SLB  = slb_p[0];          // 30
  const int TOUT = T_TOT - SLB;       // 20

  // ---- stage masked weights + folded biases (block-wide, once) ----
  for (int i = tid; i < 80 * 20; i += TPB) {
    ((float*)sWih)[i] = 0.f;
    ((float*)sWhh)[i] = 0.f;
  }
  for (int i = tid; i < 32 * 20; i += TPB) ((float*)sWlin)[i] = 0.f;
  __syncthreads();
  for (int i = tid; i < 68 * 17; i += TPB) {
    int col = i / 17, k = i % 17;
    sWih[col][k] = W_ih[i] * mask_ih[i];
    sWhh[col][k] = W_hh[i] * mask_hh[i];
  }
  for (int i = tid; i < 17 * 17; i += TPB) sWlin[i / 17][i % 17] = W_lin[i];
  if (tid < 68) sWih[tid][17] = b_ih[tid] + b_hh[tid];  // bias row (Whh row17=0)
  if (tid < 17) sWlin[tid][17] = b_lin[tid];

  // ---- zero per-wave recurrent state; plant 1.0 at k=17 (wave-local) ----
  for (int i = lane; i < 16 * 20; i += 32) {
    ((float*)sH[w])[i]   = 0.f;
    ((float*)sOut[w])[i] = 0.f;
  }
  for (int i = lane; i < 17 * 16; i += 32) ((float*)sC[w])[i] = 0.f;
  if (lane < 16) {
    sH[w][lane][17]   = 1.0f;  // head bias source; Whh row17==0 keeps gates ok
    sOut[w][lane][17] = 1.0f;  // feedback-path bias source for gates
  }
  __syncthreads();  // weights visible to all waves

  const int b0   = (blockIdx.x * WPB + w) * 16;  // batch tile base row
  const int row  = lane & 15;                    // A-fragment row (M)
  const int colq = lane & 15;                    // B/C column within tile (N)
  const int m0   = (lane >> 4) * 8;              // C-layout row offset
  const int koff = (lane >> 4) * 2;              // A/B layout K offset

  // ---- hoist loop-invariant weight B-fragments into registers ----
  v2f rWih[5][5], rWhh[5][5], rWlin[2][5];
#pragma unroll
  for (int nt = 0; nt < 5; ++nt)
#pragma unroll
    for (int kc = 0; kc < 5; ++kc) {
      rWih[nt][kc] = *(const v2f*)&sWih[nt * 16 + colq][kc * 4 + koff];
      rWhh[nt][kc] = *(const v2f*)&sWhh[nt * 16 + colq][kc * 4 + koff];
    }
#pragma unroll
  for (int nt = 0; nt < 2; ++nt)
#pragma unroll
    for (int kc = 0; kc < 5; ++kc)
      rWlin[nt][kc] = *(const v2f*)&sWlin[nt * 16 + colq][kc * 4 + koff];

  for (int t = 0; t < T_TOT; ++t) {
    // -- xi A-fragments: teacher forcing from x, else previous head output --
    v2f ax[5], ah[5];
    if (t < SLB) {
      const float* xp =
          x + (size_t)(b0 + row) * (T_TOT * D_DIM) + (size_t)t * D_DIM;
#pragma unroll
      for (int kc = 0; kc < 5; ++kc) {
        int k = kc * 4 + koff;
        v2f av;
        av.x = (k < D_DIM) ? xp[k] : 0.f;
        av.y = (k + 1 < D_DIM) ? xp[k + 1]
                               : ((k + 1 == D_DIM) ? 1.0f : 0.f);  // k17 = 1
        ax[kc] = av;
      }
    } else {
#pragma unroll
      for (int kc = 0; kc < 5; ++kc)
        ax[kc] = *(const v2f*)&sOut[w][row][kc * 4 + koff];
    }
#pragma unroll
    for (int kc = 0; kc < 5; ++kc)
      ah[kc] = *(const v2f*)&sH[w][row][kc * 4 + koff];

    // -- gates = xi@WihT + h@WhhT (+bias via k=17): zero-C start per tile --
#pragma unroll
    for (int nt = 0; nt < 5; ++nt) {
      v8f acc = {};  // lowers to inline SRC2=0
#pragma unroll
      for (int kc = 0; kc < 5; ++kc) {
        acc = __builtin_amdgcn_wmma_f32_16x16x4_f32(
            false, ax[kc], false, rWih[nt][kc], (short)0, acc, false, false);
        acc = __builtin_amdgcn_wmma_f32_16x16x4_f32(
            false, ah[kc], false, rWhh[nt][kc], (short)0, acc, false, false);
      }
      float* gp = &sGates[w][nt * 16 + colq][m0];
      ((float4*)gp)[0] = make_float4(acc[0], acc[1], acc[2], acc[3]);
      ((float4*)gp)[1] = make_float4(acc[4], acc[5], acc[6], acc[7]);
    }
    asm volatile("" ::: "memory");  // keep LDS store->load order (HW in-order)

    // -- elementwise LSTM cell over 16 rows x 17 dims (272 elems, 32 lanes) --
    for (int e = lane; e < 272; e += 32) {
      int r = e & 15, d = e >> 4;
      float ig = sGates[w][d][r];
      float fg = sGates[w][17 + d][r];
      float gg = sGates[w][34 + d][r];
      float og = sGates[w][51 + d][r];
      float cv = sC[w][d][r];
      float cn = sigm(fg) * cv + sigm(ig) * tanh_fast(gg);
      float hn = sigm(og) * tanh_fast(cn);
      sC[w][d][r] = cn;
      sH[w][r][d] = hn;
    }
    asm volatile("" ::: "memory");

    // -- linear head (bias via k=17): only needed from t = SLB-1 onward --
    if (t >= SLB - 1) {
      v2f an[5];
#pragma unroll
      for (int kc = 0; kc < 5; ++kc)
        an[kc] = *(const v2f*)&sH[w][row][kc * 4 + koff];
#pragma unroll
      for (int nt = 0; nt < 2; ++nt) {
        v8f acc = {};
#pragma unroll
        for (int kc = 0; kc < 5; ++kc)
          acc = __builtin_amdgcn_wmma_f32_16x16x4_f32(
              false, an[kc], false, rWlin[nt][kc], (short)0, acc, false, false);
        int col = nt * 16 + colq;
        if (col < D_DIM) {
#pragma unroll
          for (int m = 0; m < 8; ++m) sOut[w][m0 + m][col] = acc[m];
          if (t >= SLB) {
#pragma unroll
            for (int m = 0; m < 8; ++m)
              out[((size_t)(b0 + m0 + m) * TOUT + (t - SLB)) * D_DIM + col] =
                  acc[m];
          }
        }
      }
      asm volatile("" ::: "memory");
    }
  }
}

extern "C" void kernel_launch(void* const* d_in, const int* in_sizes, int n_in,
                              void* d_out, int out_size, void* d_ws,
                              size_t ws_size, hipStream_t stream) {
  const float* x       = (const float*)d_in[0];
  const float* W_ih    = (const float*)d_in[1];
  const float* W_hh    = (const float*)d_in[2];
  const float* b_ih    = (const float*)d_in[3];
  const float* b_hh    = (const float*)d_in[4];
  const float* W_lin   = (const float*)d_in[5];
  const float* b_lin   = (const float*)d_in[6];
  const float* mask_ih = (const float*)d_in[7];
  const float* mask_hh = (const float*)d_in[8];
  const int*   slb     = (const int*)d_in[9];
  float* out = (float*)d_out;

  dim3 grid(B_TOT / (WPB * 16));  // 1024 blocks: 4096 waves, 16 rows each
  dim3 block(TPB);
  hipLaunchKernelGGL(lstm_wmma_kernel, grid, block, 0, stream, x, W_ih, W_hh,
                     b_ih, b_hh, W_lin, b_lin, mask_ih, mask_hh, slb, out);
}